// SparseLinear_62380105007472
// MI455X (gfx1250) — compile-verified
//
#include <hip/hip_runtime.h>

// Problem constants (from reference)
#define IN_FEATURES  4096
#define OUT_FEATURES 4096
#define BATCH        512
#define NNZ_TOTAL    838860

// Tiling
#define BTILE   16                    // batch columns per scatter block
#define NBT     (BATCH / BTILE)       // 32 batch tiles
#define NCHUNK  8                     // nnz chunks -> 32*8 = 256 blocks
#define SCATTER_THREADS 512           // 16 wave32 -> 4 waves/SIMD (1 WG/WGP)
#define WAVES_PER_BLOCK (SCATTER_THREADS / 32)
#define TILE_E  64                            // COO entries per async-staged tile
#define NTILES  (NNZ_TOTAL / TILE_E)          // 13107 full tiles
#define TAIL_START (NTILES * TILE_E)          // 838848 (12-entry tail)
#define WAVE_SLOTS (NCHUNK * WAVES_PER_BLOCK) // 128 wave slots per batch tile

// ---------------------------------------------------------------------------
// Kernel 1: transpose x [BATCH, IN] -> xT [IN, BATCH] (coalesced both sides)
// ---------------------------------------------------------------------------
__global__ __launch_bounds__(256) void xpose_x(const float* __restrict__ x,
                                               float* __restrict__ xT) {
    __shared__ float tile[32][33];
    const int i0 = blockIdx.x * 32;   // feature tile
    const int b0 = blockIdx.y * 32;   // batch tile
    const int tx = threadIdx.x, ty = threadIdx.y;
#pragma unroll
    for (int k = 0; k < 32; k += 8)
        tile[ty + k][tx] = x[(long)(b0 + ty + k) * IN_FEATURES + i0 + tx];
    __syncthreads();
#pragma unroll
    for (int k = 0; k < 32; k += 8)
        xT[(long)(i0 + ty + k) * BATCH + b0 + tx] = tile[tx][ty + k];
}

// ---------------------------------------------------------------------------
// Async staging of one 64-entry COO tile into per-wave LDS buffers.
// rows: 512B (b128 per lane), cols: 512B (b128 per lane), vals: 256B (b64).
// Tracked on ASYNCcnt (3 ops per stage); each wave stages only what it
// consumes, so no cross-wave barrier is needed for the staging buffers.
// LDS byte offset = low 32 bits of the generic pointer (LDS aperture rule).
// ---------------------------------------------------------------------------
__device__ __forceinline__ void stage_tile(const long long* rows,
                                           const long long* cols,
                                           const float* vals, int t, int lane,
                                           long long* bR, long long* bC,
                                           float* bV) {
    unsigned long long gR = (unsigned long long)(const char*)(rows + t * TILE_E) + lane * 16;
    unsigned long long gC = (unsigned long long)(const char*)(cols + t * TILE_E) + lane * 16;
    unsigned long long gV = (unsigned long long)(const char*)(vals + t * TILE_E) + lane * 8;
    unsigned dR = (unsigned)(unsigned long long)((char*)bR + lane * 16);
    unsigned dC = (unsigned)(unsigned long long)((char*)bC + lane * 16);
    unsigned dV = (unsigned)(unsigned long long)((char*)bV + lane * 8);
    asm volatile("global_load_async_to_lds_b128 %0, %1, off"
                 :: "v"(dR), "v"(gR) : "memory");
    asm volatile("global_load_async_to_lds_b128 %0, %1, off"
                 :: "v"(dC), "v"(gC) : "memory");
    asm volatile("global_load_async_to_lds_b64 %0, %1, off"
                 :: "v"(dV), "v"(gV) : "memory");
}

// ---------------------------------------------------------------------------
// Kernel 2: COO scatter with full-row LDS accumulator.
//   acc[4096 rows x 16 batch cols] = 256 KB LDS -> no row filtering, no
//   divergence: every scanned entry is useful. 16-lane groups apply one entry
//   across 16 batch columns via ds_add_f32. COO stream is double-buffered into
//   LDS with GLOBAL_LOAD_ASYNC_TO_LDS + s_wait_asynccnt.
//   Flush: non-returning global f32 atomics into outT[row, b].
// ---------------------------------------------------------------------------
__global__ __launch_bounds__(SCATTER_THREADS) void coo_scatter(
    const float*     __restrict__ xT,
    const long long* __restrict__ rows,
    const long long* __restrict__ cols,
    const float*     __restrict__ vals,
    float*           __restrict__ outT) {
    __shared__ float     acc[OUT_FEATURES * BTILE];          // 256 KB
    __shared__ long long sR[WAVES_PER_BLOCK][2][TILE_E];     // 16 KB
    __shared__ long long sC[WAVES_PER_BLOCK][2][TILE_E];     // 16 KB
    __shared__ float     sV[WAVES_PER_BLOCK][2][TILE_E];     // 8 KB

    const int tid = threadIdx.x;
    for (int i = tid; i < OUT_FEATURES * BTILE; i += SCATTER_THREADS)
        acc[i] = 0.0f;
    __syncthreads();

    const int bt   = blockIdx.x;        // batch tile 0..31
    const int w    = tid >> 5;          // wave in block
    const int lane = tid & 31;
    const int sub  = lane >> 4;         // which of 2 entries this half-wave does
    const int l16  = lane & 15;         // batch column within tile
    const int bb   = bt * BTILE + l16;
    const int slot = blockIdx.y * WAVES_PER_BLOCK + w;   // 0..127

    int t = slot;
    if (t < NTILES) {
        stage_tile(rows, cols, vals, t, lane, sR[w][0], sC[w][0], sV[w][0]);
        int buf = 0;
        for (; t < NTILES; t += WAVE_SLOTS) {
            const int tn = t + WAVE_SLOTS;
            if (tn < NTILES) {
                stage_tile(rows, cols, vals, tn, lane,
                           sR[w][buf ^ 1], sC[w][buf ^ 1], sV[w][buf ^ 1]);
                // keep next tile's 3 async ops in flight; current tile done
                asm volatile("s_wait_asynccnt 3" ::: "memory");
            } else {
                asm volatile("s_wait_asynccnt 0" ::: "memory");
            }
            const long long* tR = sR[w][buf];
            const long long* tC = sC[w][buf];
            const float*     tV = sV[w][buf];
#pragma unroll 8
            for (int i = 0; i < TILE_E; i += 2) {
                const int   e  = i + sub;
                const int   r  = (int)tR[e];
                const int   c  = (int)tC[e];
                const float v  = tV[e];
                const float xv = xT[(long)c * BATCH + bb];
                atomicAdd(&acc[r * BTILE + l16], xv * v);    // ds_add_f32
            }
            buf ^= 1;
        }
    }

    // 12-entry tail handled once per batch tile (chunk 0, wave 0), direct loads
    if (blockIdx.y == 0 && w == 0) {
        for (int e = TAIL_START + sub; e < NNZ_TOTAL; e += 2) {
            const int   r  = (int)rows[e];
            const int   c  = (int)cols[e];
            const float v  = vals[e];
            const float xv = xT[(long)c * BATCH + bb];
            atomicAdd(&acc[r * BTILE + l16], xv * v);
        }
    }
    __syncthreads();

    // Flush LDS tile to global accumulator (skip exact zeros: adding 0 is a nop)
    for (int i = tid; i < OUT_FEATURES * BTILE; i += SCATTER_THREADS) {
        const float a = acc[i];
        if (a != 0.0f) {
            const int r = i >> 4;      // BTILE == 16
            const int l = i & 15;
            unsafeAtomicAdd(&outT[(long)r * BATCH + bt * BTILE + l], a);
        }
    }
}

// ---------------------------------------------------------------------------
// Kernel 3: out[b, r] = outT[r, b] + bias[r]  (tiled transpose + bias)
// ---------------------------------------------------------------------------
__global__ __launch_bounds__(256) void finalize(const float* __restrict__ outT,
                                                const float* __restrict__ bias,
                                                float* __restrict__ out) {
    __shared__ float tile[32][33];
    const int r0 = blockIdx.x * 32;
    const int b0 = blockIdx.y * 32;
    const int tx = threadIdx.x, ty = threadIdx.y;
#pragma unroll
    for (int k = 0; k < 32; k += 8)
        tile[ty + k][tx] = outT[(long)(r0 + ty + k) * BATCH + b0 + tx];
    __syncthreads();
#pragma unroll
    for (int k = 0; k < 32; k += 8)
        out[(long)(b0 + ty + k) * OUT_FEATURES + r0 + tx] =
            tile[tx][ty + k] + bias[r0 + tx];
}

// ---------------------------------------------------------------------------
extern "C" void kernel_launch(void* const* d_in, const int* in_sizes, int n_in,
                              void* d_out, int out_size, void* d_ws, size_t ws_size,
                              hipStream_t stream) {
    const float*     x    = (const float*)d_in[0];
    const long long* rows = (const long long*)d_in[1];  // int64 per reference
    const long long* cols = (const long long*)d_in[2];  // int64 per reference
    const float*     vals = (const float*)d_in[3];
    const float*     bias = (const float*)d_in[4];
    float*           out  = (float*)d_out;

    // Workspace layout: xT (8 MB) | outT (8 MB)
    float* xT   = (float*)d_ws;
    float* outT = xT + (size_t)IN_FEATURES * BATCH;

    // outT must be zeroed every call (deterministic; graph-capturable memset)
    hipMemsetAsync(outT, 0, (size_t)OUT_FEATURES * BATCH * sizeof(float), stream);

    dim3 tb(32, 8);
    xpose_x<<<dim3(IN_FEATURES / 32, BATCH / 32), tb, 0, stream>>>(x, xT);

    coo_scatter<<<dim3(NBT, NCHUNK), SCATTER_THREADS, 0, stream>>>(
        xT, rows, cols, vals, outT);

    finalize<<<dim3(OUT_FEATURES / 32, BATCH / 32), tb, 0, stream>>>(
        outT, bias, out);
}